// SpikingNeuron_30580167147909
// MI455X (gfx1250) — compile-verified
//
#include <hip/hip_runtime.h>
#include <hip/hip_bf16.h>

namespace {
constexpr int kT = 1000;
constexpr int kB = 32;
constexpr int kN = 2048;   // power of two: gid>>11 == b
constexpr int kBN = kB * kN;
constexpr int kBlock = 256;
constexpr int kRows = 8;             // t-rows per tile (1000 % 8 == 0)
constexpr int kTiles = kT / kRows;   // 125
constexpr int kWaves = kBlock / 32;  // 8 waves per block (wave32)
}

#define GLOBAL_AS __attribute__((address_space(1)))
#define LDS_AS    __attribute__((address_space(3)))

#if __has_builtin(__builtin_amdgcn_global_load_async_to_lds_b128) && \
    __has_builtin(__builtin_amdgcn_global_load_async_to_lds_b32)  && \
    __has_builtin(__builtin_amdgcn_s_wait_asynccnt)
#define HAVE_ASYNC_LDS 1
#else
#define HAVE_ASYNC_LDS 0
#endif

#if HAVE_ASYNC_LDS
// Builtin signatures (confirmed by hipcc diagnostics):
//   b128: (int4 AS1*, int4 AS3*, imm offset, imm cpol)
//   b32 : (int  AS1*, int  AS3*, imm offset, imm cpol)
typedef int v4i __attribute__((vector_size(16)));
typedef GLOBAL_AS v4i* gptr_v4i;
typedef LDS_AS    v4i* lptr_v4i;
typedef GLOBAL_AS int* gptr_i32;
typedef LDS_AS    int* lptr_i32;

__device__ __forceinline__ void async_g2l_b128(const void* g, void* l) {
  __builtin_amdgcn_global_load_async_to_lds_b128((gptr_v4i)g, (lptr_v4i)l, 0, 0);
}
__device__ __forceinline__ void async_g2l_b32(const void* g, void* l) {
  __builtin_amdgcn_global_load_async_to_lds_b32((gptr_i32)g, (lptr_i32)l, 0, 0);
}
// compiler memory barrier after the counter wait pins LDS reads behind it
#define WAIT_ASYNC(n)                         \
  do {                                        \
    __builtin_amdgcn_s_wait_asynccnt(n);      \
    asm volatile("" ::: "memory");            \
  } while (0)
// drain this wave's DS reads before the async engine may overwrite a buffer
#define FENCE_DS() asm volatile("s_wait_dscnt 0x0" ::: "memory")
#endif

// ---------------------------------------------------------------------------
// Zero the workspace accumulators (pop[T*B] + active[B]) each launch.
// ---------------------------------------------------------------------------
__global__ void zero_u32_kernel(unsigned* __restrict__ p, int n) {
  int i = blockIdx.x * blockDim.x + threadIdx.x;
  if (i < n) p[i] = 0u;
}

// ---------------------------------------------------------------------------
// Fused streaming scan. Front-end: per-wave double-buffered async global->LDS
// staging (2 x b128 per 1KB tile, s_wait_asynccnt ping-pong, no barriers).
// Back-end: branchless ISI state machine + wave32 ballot population counts.
// ---------------------------------------------------------------------------
__global__ __launch_bounds__(kBlock) void spike_scan_kernel(
    const float* __restrict__ spikes, float* __restrict__ out,
    unsigned* __restrict__ pop, unsigned* __restrict__ active) {
  __shared__ unsigned lds_pop[kT];
#if HAVE_ASYNC_LDS
  __shared__ float stage[kWaves][2][kRows][32] __attribute__((aligned(16)));
#endif
  const int tid = threadIdx.x;
  for (int i = tid; i < kT; i += kBlock) lds_pop[i] = 0u;
  __syncthreads();

  const int gid  = blockIdx.x * kBlock + tid;  // == b*kN + n ; block spans one b
  const int b    = gid >> 11;
  const int lane = tid & 31;
  const int w    = tid >> 5;
  const bool lane0 = (lane == 0);

  int   last   = -1;   // time of previous spike
  int   sum_g  = 0;    // sum of ISIs (telescopes, accumulated masked)
  int   sum_g2 = 0;    // sum of squared ISIs (<= 998001, exact)
  float totalf = 0.0f; // spikes are exactly 0.0/1.0

  auto step = [&](int t, float v) {
    const bool s = v > 0.0f;
    const unsigned m = __builtin_amdgcn_ballot_w32(s);   // pop count for this t
    if (lane0) atomicAdd(&lds_pop[t], (unsigned)__popc(m));
    const int g  = t - last;
    const int gm = (s && last >= 0) ? g : 0;
    sum_g  += gm;
    sum_g2 += __mul24(gm, gm);
    last = s ? t : last;
    totalf += v;
  };

#if HAVE_ASYNC_LDS
  // Wave-uniform global base -> SGPR so tiles advance via SALU (SADDR form).
  const unsigned wave0 = __builtin_amdgcn_readfirstlane((unsigned)(gid & ~31));
  const float* gbase = spikes + wave0;
  // Chunk c (0..63) of a tile: row = c>>3, 16B part = c&7. Lane handles
  // chunks {lane, lane+32}; offsets are loop-invariant VGPRs.
  const int c0 = lane, c1 = lane + 32;
  const unsigned off0 = (unsigned)((c0 >> 3) * kBN + (c0 & 7) * 4);
  const unsigned off1 = (unsigned)((c1 >> 3) * kBN + (c1 & 7) * 4);
  float* const l0a = &stage[w][0][0][0] + c0 * 4;  // LDS floats = c*4 (row-major)
  float* const l1a = &stage[w][0][0][0] + c1 * 4;
  float* const l0b = &stage[w][1][0][0] + c0 * 4;
  float* const l1b = &stage[w][1][0][0] + c1 * 4;

  auto issue = [&](int tile, int buf) {
    FENCE_DS();  // prior consume of this buffer fully read before overwrite
    const float* g = gbase + (unsigned)tile * (unsigned)(kRows * kBN);
    if (buf == 0) { async_g2l_b128(g + off0, l0a); async_g2l_b128(g + off1, l1a); }
    else          { async_g2l_b128(g + off0, l0b); async_g2l_b128(g + off1, l1b); }
  };
  auto consume = [&](int tile, int buf) {
    const float* s = &stage[w][buf][0][lane];
#pragma unroll
    for (int r = 0; r < kRows; ++r) step(tile * kRows + r, s[r * 32]);
  };

  issue(0, 0);
  for (int j = 0; j < (kTiles - 1) / 2; ++j) {   // 62 iterations, static bufs
    issue(2 * j + 1, 1); WAIT_ASYNC(2); consume(2 * j, 0);
    issue(2 * j + 2, 0); WAIT_ASYNC(2); consume(2 * j + 1, 1);
  }
  WAIT_ASYNC(0);
  consume(kTiles - 1, 0);                        // tile 124 -> buf 0
#else
  // Fallback: direct unrolled coalesced loads + global_prefetch_b8.
  const float* __restrict__ col = spikes + gid;
  for (int t0 = 0; t0 < kT; t0 += kRows) {
    float v[kRows];
#pragma unroll
    for (int u = 0; u < kRows; ++u) v[u] = col[(size_t)(t0 + u) * kBN];
    if (t0 + kRows + 16 <= kT) {
#pragma unroll
      for (int u = 0; u < kRows; ++u)
        __builtin_prefetch(col + (size_t)(t0 + u + 16) * kBN, 0, 1);
    }
#pragma unroll
    for (int u = 0; u < kRows; ++u) step(t0 + u, v[u]);
  }
#endif

  // ---- per-neuron outputs ----
  const float fr   = totalf / (kT * 0.001f);
  const float fcnt = fmaxf(totalf - 1.0f, 0.0f);           // n_spikes - 1
  const float mean = (float)sum_g / fmaxf(fcnt, 1.0f);
  const float var  = ((float)sum_g2 - fcnt * mean * mean) / fmaxf(fcnt - 1.0f, 1.0f);
  const float cv   = (fcnt >= 1.0f)
                       ? sqrtf(fmaxf(var, 0.0f)) / fmaxf(mean, 1e-12f)
                       : 0.0f;

  out[gid]                = fr;      // firing_rates   [0, BN)
  out[kBN + gid]          = cv;      // cv_isi         [BN, 2BN)
  out[2 * kBN + kB + gid] = totalf;  // total_spikes   [2BN+B, 3BN+B)

  const unsigned am = __builtin_amdgcn_ballot_w32(totalf > 0.0f);
  if (lane0) atomicAdd(&active[b], (unsigned)__popc(am));

  __syncthreads();
  for (int t = tid; t < kT; t += kBlock)
    atomicAdd(&pop[b * kT + t], lds_pop[t]);
}

// ---------------------------------------------------------------------------
// Synchrony: lag-1 circular autocorrelation of pop[T] per batch.
// Circular roll => sum(ym^2)==sum(xm^2); den=sqrt(A*A)=A; std>0 <=> A>0.
// ---------------------------------------------------------------------------
__device__ inline float block_reduce_256(float v, float* red) {
  const int tid = threadIdx.x;
  red[tid] = v;
  __syncthreads();
#pragma unroll
  for (int s = kBlock / 2; s > 0; s >>= 1) {
    if (tid < s) red[tid] += red[tid + s];
    __syncthreads();
  }
  const float r = red[0];
  __syncthreads();
  return r;
}

__global__ __launch_bounds__(kBlock) void spike_finalize_kernel(
    const unsigned* __restrict__ pop, const unsigned* __restrict__ active,
    float* __restrict__ out) {
  __shared__ unsigned popu[kT];
  __shared__ float red[kBlock];
  const int b = blockIdx.x;
  const int tid = threadIdx.x;

#if HAVE_ASYNC_LDS
  // Stage the 4KB pop row straight into LDS with the async engine.
  for (int t = tid; t < kT; t += kBlock)
    async_g2l_b32(&pop[b * kT + t], &popu[t]);
  WAIT_ASYNC(0);
#else
  for (int t = tid; t < kT; t += kBlock) popu[t] = pop[b * kT + t];
#endif
  __syncthreads();

  float psum = 0.0f;
  for (int t = tid; t < kT; t += kBlock) psum += (float)popu[t];
  const float m = block_reduce_256(psum, red) / (float)kT;

  float num = 0.0f, den = 0.0f;
  for (int t = tid; t < kT; t += kBlock) {
    const float x = (float)popu[t] - m;
    const int tp = (t == 0) ? (kT - 1) : (t - 1);
    const float y = (float)popu[tp] - m;
    num += x * y;
    den += x * x;
  }
  const float numT = block_reduce_256(num, red);
  const float denT = block_reduce_256(den, red);

  if (tid == 0) {
    const float syn = (denT > 0.0f) ? (numT / fmaxf(denT, 1e-12f)) : 0.0f;
    out[2 * kBN + b]      = syn;              // synchrony      [2BN, 2BN+B)
    out[3 * kBN + kB + b] = (float)active[b]; // active_neurons [3BN+B, 3BN+2B)
  }
}

// ---------------------------------------------------------------------------
extern "C" void kernel_launch(void* const* d_in, const int* in_sizes, int n_in,
                              void* d_out, int out_size, void* d_ws, size_t ws_size,
                              hipStream_t stream) {
  (void)in_sizes; (void)n_in; (void)out_size; (void)ws_size;
  const float* spikes = (const float*)d_in[0];
  float* out = (float*)d_out;

  unsigned* pop    = (unsigned*)d_ws;  // [B*T] u32
  unsigned* active = pop + kB * kT;    // [B]   u32
  const int nzero = kB * kT + kB;

  zero_u32_kernel<<<(nzero + 255) / 256, 256, 0, stream>>>(pop, nzero);
  spike_scan_kernel<<<kBN / kBlock, kBlock, 0, stream>>>(spikes, out, pop, active);
  spike_finalize_kernel<<<kB, kBlock, 0, stream>>>(pop, active, out);
}